// DGCNLayer_2345052144351
// MI455X (gfx1250) — compile-verified
//
#include <hip/hip_runtime.h>
#include <math.h>

typedef __attribute__((ext_vector_type(2))) float v2f;
typedef __attribute__((ext_vector_type(8))) float v8f;
typedef int v4i __attribute__((vector_size(16)));

#define D 128

// fast deg^-1/2 via v_rsq_f32 (~1ulp, plenty for normalization factors)
__device__ __forceinline__ float fast_rsq(float x) {
  return __builtin_amdgcn_rsqf(x);
}
__device__ __forceinline__ float fast_rcp(float x) {
  return __builtin_amdgcn_rcpf(x);
}

// ---------------------------------------------------------------------------
// 1) zero the workspace region (agg[N*128] + out_deg[N] + in_deg[N])
// ---------------------------------------------------------------------------
__global__ void dgcn_zero_kernel(unsigned* __restrict__ p, long n) {
  long i = (long)blockIdx.x * blockDim.x + threadIdx.x;
  long stride = (long)gridDim.x * blockDim.x;
  for (; i < n; i += stride) p[i] = 0u;
}

// ---------------------------------------------------------------------------
// 2) in/out degree via integer atomics
// ---------------------------------------------------------------------------
__global__ void dgcn_degree_kernel(const int* __restrict__ src,
                                   const int* __restrict__ dst,
                                   unsigned* __restrict__ out_deg,
                                   unsigned* __restrict__ in_deg, int E) {
  int e = blockIdx.x * blockDim.x + threadIdx.x;
  if (e < E) {
    atomicAdd(&out_deg[src[e]], 1u);
    atomicAdd(&in_deg[dst[e]], 1u);
  }
}

// ---------------------------------------------------------------------------
// 3) per-edge message + scatter: one wave32 per edge, float4 per lane.
//    agg[dst] += h[src] * 2^-dist * outdeg(src)^-1/2   (raw sum; norm folded later)
//    2^-dist is built exactly from the exponent bits; rsq is v_rsq_f32.
// ---------------------------------------------------------------------------
__global__ __launch_bounds__(256) void dgcn_edge_kernel(
    const float* __restrict__ h, const int* __restrict__ src,
    const int* __restrict__ dst, const int* __restrict__ dist,
    const unsigned* __restrict__ out_deg, float* __restrict__ agg, int E) {
  int e = blockIdx.x * (blockDim.x >> 5) + (threadIdx.x >> 5);
  if (e >= E) return;
  int lane = threadIdx.x & 31;
  int s = src[e];
  int d = dst[e];
  int dd = dist[e];
  float pw = __int_as_float((127 - dd) << 23);          // exact 2^-dist
  float w = pw * fast_rsq((float)out_deg[s]);
  float4 v = ((const float4*)(h + (size_t)s * D))[lane];
  float* ap = agg + (size_t)d * D + lane * 4;
  atomicAdd(ap + 0, v.x * w);
  atomicAdd(ap + 1, v.y * w);
  atomicAdd(ap + 2, v.z * w);
  atomicAdd(ap + 3, v.w * w);
}

// ---------------------------------------------------------------------------
// 4) rst = (agg @ W) * indeg^-3/2 + bias, using V_WMMA_F32_16X16X4_F32.
//    Block = 256 threads (8 waves) -> 64-row strip (4 row-tiles per wave,
//    B fragment reused across 4 WMMAs per k-step). Weight (64KB) staged in
//    LDS once per block via GLOBAL_LOAD_ASYNC_TO_LDS_B128 (ASYNCcnt-tracked).
// ---------------------------------------------------------------------------
__global__ __launch_bounds__(256) void dgcn_gemm_wmma_kernel(
    const float* __restrict__ agg, const unsigned* __restrict__ in_deg,
    const float* __restrict__ W, const float* __restrict__ bias,
    float* __restrict__ out, int N) {
  __shared__ float sW[D * D];  // 64 KB

#if __has_builtin(__builtin_amdgcn_global_load_async_to_lds_b128)
  {
    const float4* g = (const float4*)W;
    float4* l = (float4*)sW;
    for (int i = threadIdx.x; i < (D * D) / 4; i += 256) {
      __builtin_amdgcn_global_load_async_to_lds_b128(
          (__attribute__((address_space(1))) v4i*)(g + i),
          (__attribute__((address_space(3))) v4i*)(l + i),
          /*offset=*/0, /*cpol=*/0);
    }
#if __has_builtin(__builtin_amdgcn_s_wait_asynccnt)
    __builtin_amdgcn_s_wait_asynccnt(0);
#else
    asm volatile("s_wait_asynccnt 0x0" ::: "memory");
#endif
  }
#else
  for (int i = threadIdx.x; i < (D * D) / 4; i += 256)
    ((float4*)sW)[i] = ((const float4*)W)[i];
#endif
  __syncthreads();

  const int lane = threadIdx.x & 31;
  const int wave = threadIdx.x >> 5;
  const int half = lane >> 4;             // 0: K{0,1}, 1: K{2,3}
  const int n = wave * 16 + (lane & 15);  // this wave's output column
  const int rowblk = blockIdx.x * 64;
  const int row0 = rowblk + (lane & 15);

  // clamp A-row pointers for the (only) partial tail block; stores are guarded
  const float* ar0;
  const float* ar1;
  const float* ar2;
  const float* ar3;
  {
    int m0 = row0;           if (m0 >= N) m0 = N - 1;
    int m1 = row0 + 16;      if (m1 >= N) m1 = N - 1;
    int m2 = row0 + 32;      if (m2 >= N) m2 = N - 1;
    int m3 = row0 + 48;      if (m3 >= N) m3 = N - 1;
    ar0 = agg + (size_t)m0 * D;
    ar1 = agg + (size_t)m1 * D;
    ar2 = agg + (size_t)m2 * D;
    ar3 = agg + (size_t)m3 * D;
  }

  v8f c0 = {}, c1 = {}, c2 = {}, c3 = {};
  #pragma unroll 2
  for (int k = 0; k < D; k += 4) {
    const int kb = k + half * 2;  // even -> 8B-aligned v2f loads
    v2f b;
    b.x = sW[kb * D + n];
    b.y = sW[(kb + 1) * D + n];
    v2f a0 = *(const v2f*)(ar0 + kb);
    v2f a1 = *(const v2f*)(ar1 + kb);
    v2f a2 = *(const v2f*)(ar2 + kb);
    v2f a3 = *(const v2f*)(ar3 + kb);
    c0 = __builtin_amdgcn_wmma_f32_16x16x4_f32(false, a0, false, b, (short)0, c0, false, false);
    c1 = __builtin_amdgcn_wmma_f32_16x16x4_f32(false, a1, false, b, (short)0, c1, false, false);
    c2 = __builtin_amdgcn_wmma_f32_16x16x4_f32(false, a2, false, b, (short)0, c2, false, false);
    c3 = __builtin_amdgcn_wmma_f32_16x16x4_f32(false, a3, false, b, (short)0, c3, false, false);
  }

  // C/D layout: element j of the accumulator -> row tilebase + j + 8*half,
  // column n. Fold agg/indeg mean + symmetric norm as indeg^-3/2.
  const float bn = bias[n];
  auto store_tile = [&](v8f c, int tilebase) {
    #pragma unroll
    for (int j = 0; j < 8; ++j) {
      int r = tilebase + half * 8 + j;
      if (r < N) {
        float id = (float)in_deg[r];
        float sc = fast_rsq(id) * fast_rcp(id);  // indeg^-3/2
        out[(size_t)r * D + n] = c[j] * sc + bn;
      }
    }
  };
  store_tile(c0, rowblk);
  store_tile(c1, rowblk + 16);
  store_tile(c2, rowblk + 32);
  store_tile(c3, rowblk + 48);
}

// ---------------------------------------------------------------------------
extern "C" void kernel_launch(void* const* d_in, const int* in_sizes, int n_in,
                              void* d_out, int out_size, void* d_ws, size_t ws_size,
                              hipStream_t stream) {
  const float* h    = (const float*)d_in[0];
  const int*   src  = (const int*)d_in[1];
  const int*   dst  = (const int*)d_in[2];
  const int*   dist = (const int*)d_in[3];
  const float* W    = (const float*)d_in[4];
  const float* bias = (const float*)d_in[5];
  float* out = (float*)d_out;

  const int N = in_sizes[0] / D;
  const int E = in_sizes[1];

  // workspace layout: agg[N*D] floats, then out_deg[N], in_deg[N] uints
  float* agg = (float*)d_ws;
  unsigned* out_deg = (unsigned*)(agg + (size_t)N * D);
  unsigned* in_deg = out_deg + N;

  const long zwords = (long)N * D + 2L * N;
  dgcn_zero_kernel<<<2048, 256, 0, stream>>>((unsigned*)d_ws, zwords);

  dgcn_degree_kernel<<<(E + 255) / 256, 256, 0, stream>>>(src, dst, out_deg, in_deg, E);

  // one wave (32 lanes) per edge; 8 edges per 256-thread block
  dgcn_edge_kernel<<<(E + 7) / 8, 256, 0, stream>>>(h, src, dst, dist, out_deg, agg, E);

  dgcn_gemm_wmma_kernel<<<(N + 63) / 64, 256, 0, stream>>>(agg, in_deg, W, bias, out, N);
}